// GATLayer_52664888983979
// MI455X (gfx1250) — compile-verified
//
#include <hip/hip_runtime.h>
#include <hip/hip_bf16.h>

typedef __attribute__((ext_vector_type(2))) float v2f;
typedef __attribute__((ext_vector_type(8))) float v8f;

#define NHEADS 4
#define ODIM   16
#define HDIM   64   // NHEADS*ODIM
#define KDIM   128

__device__ __forceinline__ float lrelu(float x) { return x > 0.f ? x : 0.01f * x; }

// order-preserving float<->uint encoding for atomicMax-based segment max
__device__ __forceinline__ unsigned fenc(float f) {
  unsigned u = __float_as_uint(f);
  return (u & 0x80000000u) ? ~u : (u | 0x80000000u);
}
__device__ __forceinline__ float fdec(unsigned k) {
  return (k & 0x80000000u) ? __uint_as_float(k & 0x7FFFFFFFu) : __uint_as_float(~k);
}

// ---------------------------------------------------------------- init
__global__ void k_init(float* __restrict__ dout, unsigned* __restrict__ mkey,
                       float* __restrict__ ssum, float* __restrict__ Ssum, int Nn) {
  int i = blockIdx.x * blockDim.x + threadIdx.x;
  if (i < Nn * HDIM) dout[i] = 0.f;
  if (i < Nn * NHEADS) { mkey[i] = 0x007FFFFFu; /* enc(-inf) */ ssum[i] = 0.f; }
  if (i == 0) *Ssum = 0.f;
}

// ---------------------------------------------------------------- GEMM: h = nodes @ W^T + b
// One 16x16 output tile per wave; block = 4 waves = the 4 column tiles (HDIM=64).
// V_WMMA_F32_16X16X4_F32: A 16x4 (2 VGPR/lane, lane L: M=L%16, K=(L/16)*2+v),
// B 4x16 (lane L: N=L%16, K=(L/16)*2+v), C/D 16x16 (lane L, v: M=(L/16)*8+v, N=L%16).
__global__ void k_gemm(const float* __restrict__ nodes, const float* __restrict__ Wm,
                       const float* __restrict__ bias, float* __restrict__ hbuf, int Nn) {
  const int wave   = threadIdx.x >> 5;
  const int lane   = threadIdx.x & 31;
  const int half   = lane >> 4;
  const int l16    = lane & 15;
  const int m_base = blockIdx.x * 16;          // Nn % 16 == 0 (50000/16 = 3125)
  const int n_base = wave * 16;

  const float* Arow = nodes + (size_t)(m_base + l16) * KDIM + half * 2;
  const float* Brow = Wm    + (size_t)(n_base + l16) * KDIM + half * 2;  // B[k][n] = W[n][k]

  v8f c = {};
#pragma unroll
  for (int k0 = 0; k0 < KDIM; k0 += 4) {
    v2f a = *(const v2f*)(Arow + k0);
    v2f b = *(const v2f*)(Brow + k0);
    c = __builtin_amdgcn_wmma_f32_16x16x4_f32(false, a, false, b, (short)0, c,
                                              false, false);
  }

  const float bn = bias[n_base + l16];
#pragma unroll
  for (int v = 0; v < 8; ++v) {
    int row = m_base + half * 8 + v;
    hbuf[(size_t)row * HDIM + n_base + l16] = c[v] + bn;
  }
}

// ---------------------------------------------------------------- S = sum(edges[senders])
__global__ void k_sumS(const float* __restrict__ eattr, const int* __restrict__ snd,
                       int E, float* __restrict__ Ssum) {
  __shared__ float sh[256];
  float acc = 0.f;
  for (int i = blockIdx.x * blockDim.x + threadIdx.x; i < E; i += gridDim.x * blockDim.x)
    acc += eattr[snd[i]];
  sh[threadIdx.x] = acc;
  __syncthreads();
  for (int s = 128; s > 0; s >>= 1) {
    if ((int)threadIdx.x < s) sh[threadIdx.x] += sh[threadIdx.x + s];
    __syncthreads();
  }
  if (threadIdx.x == 0) atomicAdd(Ssum, sh[0]);
}

// ---------------------------------------------------------------- per-(edge,head) logits + segment max
__global__ void k_logits(const float* __restrict__ hbuf, const float* __restrict__ eattr,
                         const int* __restrict__ snd, const int* __restrict__ rcv,
                         const float* __restrict__ aW, const float* __restrict__ ab,
                         const float* __restrict__ Ssum,
                         float* __restrict__ lbuf, unsigned* __restrict__ mkey, int E) {
  int i = blockIdx.x * blockDim.x + threadIdx.x;
  if (i >= E * NHEADS) return;
  const int e  = i >> 2;
  const int hh = i & 3;
  const int s  = snd[e];
  const int r  = rcv[e];
  const float* hs = hbuf + (size_t)s * HDIM + hh * ODIM;
  const float* hr = hbuf + (size_t)r * HDIM + hh * ODIM;
  const float* w  = aW + hh * (2 * ODIM + 1);
  __builtin_prefetch(hs, 0, 3);
  __builtin_prefetch(hr, 0, 3);
  float acc = ab[hh];
#pragma unroll
  for (int d = 0; d < ODIM; ++d) acc += hs[d] * w[d];
#pragma unroll
  for (int d = 0; d < ODIM; ++d) acc += hr[d] * w[ODIM + d];
  acc += eattr[s] * w[2 * ODIM];
  acc = lrelu(acc) * (*Ssum);
  lbuf[i] = acc;
  atomicMax(&mkey[r * NHEADS + hh], fenc(acc));
}

// ---------------------------------------------------------------- e = exp(l - m), segment sum
__global__ void k_exp(const int* __restrict__ rcv, const unsigned* __restrict__ mkey,
                      float* __restrict__ lbuf, float* __restrict__ ssum, int E) {
  int i = blockIdx.x * blockDim.x + threadIdx.x;
  if (i >= E * NHEADS) return;
  const int e  = i >> 2;
  const int hh = i & 3;
  const int r  = rcv[e];
  float ex = __expf(lbuf[i] - fdec(mkey[r * NHEADS + hh]));
  lbuf[i] = ex;
  atomicAdd(&ssum[r * NHEADS + hh], ex);
}

// ---------------------------------------------------------------- msg scatter-add
__global__ void k_scatter(const float* __restrict__ hbuf, const int* __restrict__ snd,
                          const int* __restrict__ rcv, const float* __restrict__ lbuf,
                          const float* __restrict__ ssum, float* __restrict__ dout, int E) {
  int i = blockIdx.x * blockDim.x + threadIdx.x;
  if (i >= E * NHEADS) return;
  const int e  = i >> 2;
  const int hh = i & 3;
  const int s  = snd[e];
  const int r  = rcv[e];
  const float att = lbuf[i] / ssum[r * NHEADS + hh];
  const float* hs = hbuf + (size_t)s * HDIM + hh * ODIM;
  float* o = dout + (size_t)r * HDIM + hh * ODIM;
#pragma unroll
  for (int d = 0; d < ODIM; ++d) atomicAdd(&o[d], hs[d] * att);
}

// ---------------------------------------------------------------- final leaky relu
__global__ void k_final(float* __restrict__ dout, int n) {
  int i = blockIdx.x * blockDim.x + threadIdx.x;
  if (i < n) dout[i] = lrelu(dout[i]);
}

extern "C" void kernel_launch(void* const* d_in, const int* in_sizes, int n_in,
                              void* d_out, int out_size, void* d_ws, size_t ws_size,
                              hipStream_t stream) {
  const float* nodes = (const float*)d_in[0];
  const float* eattr = (const float*)d_in[1];
  const int*   snd   = (const int*)d_in[2];
  const int*   rcv   = (const int*)d_in[3];
  const float* Wm    = (const float*)d_in[4];
  const float* bias  = (const float*)d_in[5];
  const float* aW    = (const float*)d_in[6];
  const float* ab    = (const float*)d_in[7];
  const int Nn = in_sizes[1];      // edges is [N,1] -> N nodes
  const int E  = in_sizes[2];      // senders length
  float* dout = (float*)d_out;

  // workspace layout (~27 MB)
  float*    hbuf = (float*)d_ws;                              // N*64
  float*    lbuf = hbuf + (size_t)Nn * HDIM;                  // E*4
  unsigned* mkey = (unsigned*)(lbuf + (size_t)E * NHEADS);    // N*4
  float*    ssum = (float*)(mkey + (size_t)Nn * NHEADS);      // N*4
  float*    Ssum = ssum + (size_t)Nn * NHEADS;                // 1

  const int T  = 256;
  const int NO = Nn * HDIM;
  const int EH = E * NHEADS;

  k_init<<<(NO + T - 1) / T, T, 0, stream>>>(dout, mkey, ssum, Ssum, Nn);
  k_gemm<<<Nn / 16, 128, 0, stream>>>(nodes, Wm, bias, hbuf, Nn);
  k_sumS<<<256, T, 0, stream>>>(eattr, snd, E, Ssum);
  k_logits<<<(EH + T - 1) / T, T, 0, stream>>>(hbuf, eattr, snd, rcv, aW, ab, Ssum,
                                               lbuf, mkey, E);
  k_exp<<<(EH + T - 1) / T, T, 0, stream>>>(rcv, mkey, lbuf, ssum, E);
  k_scatter<<<(EH + T - 1) / T, T, 0, stream>>>(hbuf, snd, rcv, lbuf, ssum, dout, E);
  k_final<<<(NO + T - 1) / T, T, 0, stream>>>(dout, NO);
}